// OneLayerRGCN_62929860821517
// MI455X (gfx1250) — compile-verified
//
#include <hip/hip_runtime.h>
#include <hip/hip_bf16.h>

// RGCN forward for MI455X (gfx1250, wave32).
// out[i] = x[i]@root + bias + sum_r mean_{j in N_r(i)} x[j] @ W_r
// - dense matmuls: V_WMMA_F32_16X16X4_F32 (f32-native matrix path)
// - weight matrix staged into LDS per block via TENSOR_LOAD_TO_LDS (TDM),
//   B-fragments then come from ds_load; A-fragments stream from global.
// - per-edge scatter: float4 gather + global_atomic_add_f32
// - mean folded into GEMM epilogue: (sum/c)@W == (sum@W)/c (c is a row scalar)

typedef float v2f __attribute__((ext_vector_type(2)));
typedef float v8f __attribute__((ext_vector_type(8)));
typedef unsigned int v4u __attribute__((ext_vector_type(4)));
typedef int v4i __attribute__((ext_vector_type(4)));
typedef int v8i __attribute__((ext_vector_type(8)));

#define IN_CH 128
#define OUT_CH 128
#define N_REL 8

// ---------------------------------------------------------------- zero fill
__global__ void rgcn_zero_kernel(unsigned int* __restrict__ p, size_t n) {
    size_t i = (size_t)blockIdx.x * blockDim.x + threadIdx.x;
    size_t stride = (size_t)gridDim.x * blockDim.x;
    for (; i < n; i += stride) p[i] = 0u;
}

// ------------------------------------------------------- per-(rel,dst) count
__global__ void rgcn_count_kernel(const int* __restrict__ dst,
                                  const int* __restrict__ etype,
                                  int* __restrict__ cnt,
                                  int nedges, int nnodes) {
    int e = blockIdx.x * blockDim.x + threadIdx.x;
    if (e < nedges) {
        atomicAdd(&cnt[(size_t)etype[e] * nnodes + dst[e]], 1);
    }
}

// ---------------------------------------------- scatter x[src] -> agg[dst]
// one wave (32 lanes) per edge; 4 floats per lane (128 ch / 32 lanes)
__global__ __launch_bounds__(256)
void rgcn_scatter_kernel(const float* __restrict__ x,
                         const int* __restrict__ src,
                         const int* __restrict__ dst,
                         const int* __restrict__ etype,
                         float* __restrict__ agg,
                         int rel, int nedges) {
    int lane = threadIdx.x & 31;
    int wid  = blockIdx.x * (blockDim.x >> 5) + (threadIdx.x >> 5);
    if (wid >= nedges) return;
    if (etype[wid] != rel) return;
    int s = src[wid];
    int d = dst[wid];
    const float4 v = *(const float4*)(x + (size_t)s * IN_CH + lane * 4);
    float* o = agg + (size_t)d * IN_CH + lane * 4;
    atomicAdd(o + 0, v.x);
    atomicAdd(o + 1, v.y);
    atomicAdd(o + 2, v.z);
    atomicAdd(o + 3, v.w);
}

// -------------------------------------------------- TDM: 128x128 f32 -> LDS
// D# per CDNA5 ISA:
//  group0: count=1 | lds_addr[63:32] | global_addr[120:64] | type=2[127:126]
//  group1: data_size=2(4B)[17:16] | tensor_dim0=128[79:48] |
//          tensor_dim1=128[111:80] | tile_dim0=128[127:112] |
//          tile_dim1=128[143:128] | tensor_dim0_stride=128[207:160]
//  groups 2/3: zero (<=2D tensor)
__device__ __forceinline__
void rgcn_tdm_load_w(float* lds_dst, const float* gsrc) {
#if __has_builtin(__builtin_amdgcn_tensor_load_to_lds)
    unsigned lds_addr = (unsigned)(uintptr_t)lds_dst;   // low 32b = LDS offset
    unsigned long long ga = (unsigned long long)(uintptr_t)gsrc;
    v4u g0;
    g0.x = 1u;                                   // count=1 user descriptor
    g0.y = lds_addr;                             // lds_addr
    g0.z = (unsigned)(ga & 0xFFFFFFFFu);         // global_addr[31:0]
    g0.w = (unsigned)((ga >> 32) & 0x01FFFFFFu)  // global_addr[56:32]
         | (2u << 30);                           // type=2 ("image")
    v8i g1;
    g1[0] = (int)(2u << 16);                     // data_size=4B, wg_mask=0
    g1[1] = (int)(128u << 16);                   // tensor_dim0 = 128 (lo16)
    g1[2] = (int)(128u << 16);                   // tensor_dim1 = 128 (lo16)
    g1[3] = (int)(128u << 16);                   // tile_dim0 = 128
    g1[4] = 128;                                 // tile_dim1 = 128
    g1[5] = 128;                                 // tensor_dim0_stride = 128
    g1[6] = 0;
    g1[7] = 0;
    v4i gz = {0, 0, 0, 0};
#if __clang_major__ >= 23
    v8i gz8 = {0, 0, 0, 0, 0, 0, 0, 0};
    __builtin_amdgcn_tensor_load_to_lds(g0, g1, gz, gz, gz8, 0);
#else
    __builtin_amdgcn_tensor_load_to_lds(g0, g1, gz, gz, 0);
#endif
#if __has_builtin(__builtin_amdgcn_s_wait_tensorcnt)
    __builtin_amdgcn_s_wait_tensorcnt(0);
#endif
#endif
}

// ----------------------------------------------------------- WMMA GEMM core
// One wave computes one 16x16 tile of (A[N,128] @ B[128,128]) with
// V_WMMA_F32_16X16X4_F32; K unrolled 128/4 = 32 steps; B read from LDS.
// A layout (16x4 f32): lanes 0-15 row M=lane, K={kb,kb+1};
//                      lanes 16-31 row M=lane-16, K={kb+2,kb+3}.
// B layout (4x16 f32): lanes 0-15 col N=lane, K={kb,kb+1};
//                      lanes 16-31 col N=lane-16, K={kb+2,kb+3}.
// C/D (16x16 f32): vgpr v -> (M=v+8*hi, N=l16).
__device__ __forceinline__
v8f rgcn_tile_gemm(const float* __restrict__ A, const float* __restrict__ sW,
                   int rowbase, int colbase, int nrows) {
    const int lane = threadIdx.x & 31;
    const int hi   = lane >> 4;
    const int l16  = lane & 15;

    int arow = rowbase + l16;
    if (arow >= nrows) arow = nrows - 1;      // clamp; epilogue guards stores
    const float* aptr = A + (size_t)arow * IN_CH + 2 * hi;
    const float* bptr = sW + (size_t)(2 * hi) * OUT_CH + colbase + l16;

    v8f acc = {};
#pragma unroll
    for (int kb = 0; kb < IN_CH; kb += 4) {
        v2f a = *(const v2f*)(aptr + kb);     // one global_load_b64
        v2f b;
        b.x = bptr[(size_t)kb * OUT_CH];      // ds_load_b32
        b.y = bptr[(size_t)(kb + 1) * OUT_CH];
        acc = __builtin_amdgcn_wmma_f32_16x16x4_f32(
            /*neg_a=*/false, a, /*neg_b=*/false, b,
            /*c_mod=*/(short)0, acc, /*reuse_a=*/false, /*reuse_b=*/false);
    }
    return acc;
}

// stage the 64KB weight matrix into LDS (TDM from wave 0, else coop copy)
__device__ __forceinline__
void rgcn_stage_w(float* sW, const float* W) {
#if __has_builtin(__builtin_amdgcn_tensor_load_to_lds)
    if (threadIdx.x < 32) {                  // tensor ops are wave-level
        rgcn_tdm_load_w(sW, W);
    }
#else
    for (int i = threadIdx.x; i < IN_CH * OUT_CH / 4; i += blockDim.x)
        ((float4*)sW)[i] = ((const float4*)W)[i];
#endif
    __syncthreads();
}

// out = x @ root + bias      (initializes out)
__global__ __launch_bounds__(256)
void rgcn_root_gemm_kernel(const float* __restrict__ x,
                           const float* __restrict__ root,
                           const float* __restrict__ bias,
                           float* __restrict__ out, int nnodes) {
    __shared__ float sW[IN_CH * OUT_CH];     // 64 KB of the 320 KB WGP LDS
    rgcn_stage_w(sW, root);

    const int lane = threadIdx.x & 31;
    const int hi   = lane >> 4;
    const int l16  = lane & 15;
    const int rowbase = blockIdx.x * 16;
    const int colbase = (threadIdx.x >> 5) * 16;   // 8 waves -> 8 col tiles

    v8f acc = rgcn_tile_gemm(x, sW, rowbase, colbase, nnodes);

    const int col = colbase + l16;
    const float bv = bias[col];
#pragma unroll
    for (int v = 0; v < 8; ++v) {
        int row = rowbase + v + 8 * hi;
        if (row < nnodes) out[(size_t)row * OUT_CH + col] = acc[v] + bv;
    }
}

// out += (agg @ W_r) / max(cnt_r[row], 1)
__global__ __launch_bounds__(256)
void rgcn_rel_gemm_kernel(const float* __restrict__ agg,
                          const float* __restrict__ W,
                          const int* __restrict__ cnt,   // this relation's counts
                          float* __restrict__ out, int nnodes) {
    __shared__ float sW[IN_CH * OUT_CH];
    rgcn_stage_w(sW, W);

    const int lane = threadIdx.x & 31;
    const int hi   = lane >> 4;
    const int l16  = lane & 15;
    const int rowbase = blockIdx.x * 16;
    const int colbase = (threadIdx.x >> 5) * 16;

    v8f acc = rgcn_tile_gemm(agg, sW, rowbase, colbase, nnodes);

    const int col = colbase + l16;
#pragma unroll
    for (int v = 0; v < 8; ++v) {
        int row = rowbase + v + 8 * hi;
        if (row < nnodes) {
            int c = cnt[row];
            float inv = (c > 0) ? (1.0f / (float)c) : 1.0f;
            out[(size_t)row * OUT_CH + col] += acc[v] * inv;
        }
    }
}

// ---------------------------------------------------------------- launcher
extern "C" void kernel_launch(void* const* d_in, const int* in_sizes, int n_in,
                              void* d_out, int out_size, void* d_ws, size_t ws_size,
                              hipStream_t stream) {
    const float* x          = (const float*)d_in[0];   // [N,128]
    const int*   edge_index = (const int*)d_in[1];     // [2,E] flat
    const int*   edge_type  = (const int*)d_in[2];     // [E]
    const float* weight     = (const float*)d_in[3];   // [R,128,128]
    const float* root       = (const float*)d_in[4];   // [128,128]
    const float* bias       = (const float*)d_in[5];   // [128]
    float*       out        = (float*)d_out;           // [N,128]

    const int E = in_sizes[2];
    const int N = in_sizes[0] / IN_CH;
    const int* src = edge_index;
    const int* dst = edge_index + E;

    // workspace: cnt[R*N] ints, then agg[N*128] floats (256B aligned)
    int* cnt = (int*)d_ws;
    size_t cnt_bytes = ((size_t)N_REL * N * sizeof(int) + 255) & ~(size_t)255;
    float* agg = (float*)((char*)d_ws + cnt_bytes);

    const int row_tiles = (N + 15) / 16;     // 6250 blocks, 8 waves = 8 col tiles

    // 1. zero counts, then count per (relation, dst)
    rgcn_zero_kernel<<<2048, 256, 0, stream>>>((unsigned int*)cnt, (size_t)N_REL * N);
    rgcn_count_kernel<<<(E + 255) / 256, 256, 0, stream>>>(dst, edge_type, cnt, E, N);

    // 2. out = x @ root + bias
    rgcn_root_gemm_kernel<<<row_tiles, 256, 0, stream>>>(x, root, bias, out, N);

    // 3. per relation: zero agg -> scatter -> GEMM-accumulate with mean scale
    for (int r = 0; r < N_REL; ++r) {
        rgcn_zero_kernel<<<4096, 256, 0, stream>>>((unsigned int*)agg,
                                                   (size_t)N * IN_CH);
        rgcn_scatter_kernel<<<(E + 7) / 8, 256, 0, stream>>>(x, src, dst, edge_type,
                                                             agg, r, E);
        rgcn_rel_gemm_kernel<<<row_tiles, 256, 0, stream>>>(
            agg, weight + (size_t)r * IN_CH * OUT_CH, cnt + (size_t)r * N, out, N);
    }
}